// TFLOPLoss_14431090114867
// MI455X (gfx1250) — compile-verified
//
#include <hip/hip_runtime.h>
#include <hip/hip_bf16.h>
#include <math.h>

// Problem constants (match reference)
constexpr int kB = 8;
constexpr int kL = 3072;
constexpr int kV = 2048;
constexpr int kN = 1024;
constexpr int kM = 4;
constexpr float kNEG = -1e9f;
constexpr float kTempInv = 10.0f;   // 1 / 0.1

typedef __attribute__((ext_vector_type(2))) float v2f;
typedef __attribute__((ext_vector_type(8))) float v8f;

// ---------------------------------------------------------------- reductions
__device__ __forceinline__ float blockReduce(float v, bool isMax, float* sh) {
  #pragma unroll
  for (int o = 16; o > 0; o >>= 1) {
    float t = __shfl_xor(v, o, 32);
    v = isMax ? fmaxf(v, t) : (v + t);
  }
  const int wid = threadIdx.x >> 5, lid = threadIdx.x & 31;
  if (lid == 0) sh[wid] = v;
  __syncthreads();
  const int nw = (blockDim.x + 31) >> 5;
  if (wid == 0) {
    float r = (lid < nw) ? sh[lid] : (isMax ? -INFINITY : 0.0f);
    #pragma unroll
    for (int o = 16; o > 0; o >>= 1) {
      float t = __shfl_xor(r, o, 32);
      r = isMax ? fmaxf(r, t) : (r + t);
    }
    if (lid == 0) sh[0] = r;
  }
  __syncthreads();
  float out = sh[0];
  __syncthreads();          // allow sh reuse by a following call
  return out;
}

// ---------------------------------------------------------------- init / scatter
__global__ void init_kernel(float* __restrict__ ws) {
  int i = blockIdx.x * blockDim.x + threadIdx.x;
  int total = 16 + kB * kL;
  if (i < total) ws[i] = (i < 16) ? 0.0f : 1.0f;   // labels start at 1.0
}

__global__ void scatter_kernel(const int* __restrict__ box, float* __restrict__ label) {
  int i = blockIdx.x * blockDim.x + threadIdx.x;   // over B*N*M
  if (i >= kB * kN * kM) return;
  int idx = box[i];
  if (idx != -1) {
    int b = i / (kN * kM);
    label[b * kL + idx] = 0.0f;                    // idempotent, race-free
  }
}

// ---------------------------------------------------------------- cls loss
__global__ __launch_bounds__(256) void cls_kernel(const float* __restrict__ tag,
                                                  const int* __restrict__ tokens,
                                                  float* __restrict__ ws) {
  __shared__ float sh[8];
  const int row = blockIdx.x;                      // B*L rows, V=2048 each
  const float* x = tag + (size_t)row * kV;
  const int tid = threadIdx.x;                     // 256 threads -> 8 elems each
  float4 a = ((const float4*)x)[tid * 2];
  float4 b = ((const float4*)x)[tid * 2 + 1];
  float m = fmaxf(fmaxf(fmaxf(a.x, a.y), fmaxf(a.z, a.w)),
                  fmaxf(fmaxf(b.x, b.y), fmaxf(b.z, b.w)));
  m = blockReduce(m, true, sh);
  float s = expf(a.x - m) + expf(a.y - m) + expf(a.z - m) + expf(a.w - m)
          + expf(b.x - m) + expf(b.y - m) + expf(b.z - m) + expf(b.w - m);
  s = blockReduce(s, false, sh);
  if (tid == 0) {
    int t = tokens[row];
    float xt = x[t];
    float nll = -(xt - m - logf(s));
    if (t >= 4) {                                  // not in SPECIAL_IDS {0,1,2,3}
      atomicAdd(&ws[0], nll);
      atomicAdd(&ws[1], 1.0f);
    }
  }
}

// ---------------------------------------------------------------- pointer loss
__global__ __launch_bounds__(256) void ptr_kernel(const float* __restrict__ pl,
                                                  const int* __restrict__ box,
                                                  const int* __restrict__ mask,
                                                  float* __restrict__ ws) {
  __shared__ float sh[8];
  const int r = blockIdx.x;                        // b*N + n, L=3072 each
  const int b = r / kN;
  const float* x = pl + (size_t)r * kL;
  const int* mk = mask + (size_t)b * kL;
  const int tid = threadIdx.x;                     // 12 elems per thread
  float vals[12];
  #pragma unroll
  for (int j = 0; j < 3; ++j) {
    float4 v = ((const float4*)x)[tid * 3 + j];
    int c = tid * 12 + j * 4;
    vals[j * 4 + 0] = mk[c + 0] ? v.x : kNEG;
    vals[j * 4 + 1] = mk[c + 1] ? v.y : kNEG;
    vals[j * 4 + 2] = mk[c + 2] ? v.z : kNEG;
    vals[j * 4 + 3] = mk[c + 3] ? v.w : kNEG;
  }
  float m = -INFINITY;
  #pragma unroll
  for (int j = 0; j < 12; ++j) m = fmaxf(m, vals[j]);
  m = blockReduce(m, true, sh);
  float s = 0.0f;
  #pragma unroll
  for (int j = 0; j < 12; ++j) s += expf(vals[j] - m);
  s = blockReduce(s, false, sh);
  if (tid == 0) {
    float lse = logf(s);
    float tot = 0.0f, cnt = 0.0f;
    #pragma unroll
    for (int mi = 0; mi < kM; ++mi) {
      int idx = box[r * kM + mi];
      if (idx != -1) {
        float xl = mk[idx] ? x[idx] : kNEG;
        tot += -(xl - m - lse);
        cnt += 1.0f;
      }
    }
    if (cnt > 0.0f) {
      atomicAdd(&ws[2], tot);
      atomicAdd(&ws[3], cnt);
    }
  }
}

// ---------------------------------------------------------------- empty BCE loss
__global__ __launch_bounds__(256) void empty_kernel(const float* __restrict__ el,
                                                    const int* __restrict__ mask,
                                                    const float* __restrict__ label,
                                                    float* __restrict__ ws) {
  __shared__ float sh[8];
  int i = blockIdx.x * 256 + threadIdx.x;          // B*L divisible by 256
  float x = fminf(fmaxf(el[i], -100.0f), 100.0f);
  float lab = label[i];
  float mk = mask[i] ? 1.0f : 0.0f;
  float bce = fmaxf(x, 0.0f) - x * lab + log1pf(expf(-fabsf(x)));
  float sv = blockReduce(bce * mk, false, sh);
  float sm = blockReduce(mk, false, sh);
  if (threadIdx.x == 0) {
    atomicAdd(&ws[4], sv);
    atomicAdd(&ws[5], sm);
  }
}

// ---------------------------------------------------------------- span contrastive
// One wave (32 lanes) handles 16 rows of the [N,N] sim matrix.
// V_WMMA_F32_16X16X4_F32 with B = ones(4x16) is an exact f32 row-sum engine:
// D[i][j] = sum_k A[i][k] + C[i][j]. A layout: lanes 0-15 hold K=0,1; lanes
// 16-31 hold K=2,3 -> each lane loads a float2 of its row.
//
// Fully branchless inner loops: the diagonal is masked by feeding -inf into
// expf (which lowers to an underflow cndmask, no EXEC divergence), so EXEC
// stays all-1s through the WMMA stream as the ISA requires, and loads stay
// as clean b64s.
__global__ __launch_bounds__(32) void span_kernel(const float* __restrict__ sim,
                                                  const float* __restrict__ coef,
                                                  float* __restrict__ acc) {
  __shared__ int s_neg[16];
  const int lane = threadIdx.x;
  const int half = lane >> 4;
  const int rloc = lane & 15;
  const int b = blockIdx.x >> 6;                   // N/16 = 64 tiles per batch
  const int row0 = (blockIdx.x & 63) << 4;
  const int row = row0 + rloc;
  const float* srow = sim + ((size_t)b * kN + row) * kN;
  const float* crow = coef + ((size_t)b * kN + row) * kN;

  // pass 1: off-diagonal max of s/TEMP and negative count (2 lanes per row)
  float m = -INFINITY;
  int negc = 0;
  for (int cb = 0; cb < kN; cb += 4) {
    const int c0 = cb + half * 2;
    float2 sv = *(const float2*)(srow + c0);
    float2 cv = *(const float2*)(crow + c0);
    const bool off0 = (c0 != row);
    const bool off1 = (c0 + 1 != row);
    m = fmaxf(m, off0 ? sv.x * kTempInv : -INFINITY);
    m = fmaxf(m, off1 ? sv.y * kTempInv : -INFINITY);
    negc += (off0 && !(cv.x > 0.0f)) ? 1 : 0;
    negc += (off1 && !(cv.y > 0.0f)) ? 1 : 0;
  }
  m = fmaxf(m, __shfl_xor(m, 16, 32));
  negc += __shfl_xor(negc, 16, 32);
  if (lane < 16) s_neg[lane] = negc;
  __syncthreads();

  // pass 2: WMMA-accumulated row sums (uniform control flow, EXEC all-1s)
  v8f accP = {};
  v8f accN = {};
  v8f accC = {};
  const v2f ones = {1.0f, 1.0f};
  for (int cb = 0; cb < kN; cb += 4) {
    const int c0 = cb + half * 2;
    float2 sv = *(const float2*)(srow + c0);
    float2 cv = *(const float2*)(crow + c0);
    const bool off0 = (c0 != row);
    const bool off1 = (c0 + 1 != row);
    // Diagonal -> -inf -> expf underflows to exactly 0 (branchless cndmask).
    // Off-diagonal args are <= 0 by construction of m, so no overflow path.
    const float a0 = off0 ? (sv.x * kTempInv - m) : -INFINITY;
    const float a1 = off1 ? (sv.y * kTempInv - m) : -INFINITY;
    const float e0 = expf(a0);
    const float e1 = expf(a1);
    const bool p0 = cv.x > 0.0f;
    const bool p1 = cv.y > 0.0f;
    v2f aP = {p0 ? e0 : 0.0f, p1 ? e1 : 0.0f};
    v2f aN = {p0 ? 0.0f : e0, p1 ? 0.0f : e1};
    v2f aC = {(off0 && p0) ? cv.x : 0.0f, (off1 && p1) ? cv.y : 0.0f};
    accP = __builtin_amdgcn_wmma_f32_16x16x4_f32(false, aP, false, ones,
                                                 (short)0, accP, false, false);
    accN = __builtin_amdgcn_wmma_f32_16x16x4_f32(false, aN, false, ones,
                                                 (short)0, accN, false, false);
    accC = __builtin_amdgcn_wmma_f32_16x16x4_f32(false, aC, false, ones,
                                                 (short)0, accC, false, false);
  }

  // D layout: lanes 0-15 component v = row0+v ; lanes 16-31 component v = row0+8+v
  if (lane == 0 || lane == 16) {
    const int rbase = (lane == 0) ? 0 : 8;
    float lsum = 0.0f, lcnt = 0.0f;
    #pragma unroll
    for (int v = 0; v < 8; ++v) {
      float pos_sum = accP[v];
      float neg_sum = accN[v];
      float cp = accC[v];
      bool valid = (cp > 0.0f) && (s_neg[rbase + v] > 0);
      float cp_s = valid ? cp : 1.0f;
      float pos_s = valid ? pos_sum : 1.0f;
      float neg_s = valid ? neg_sum : 1.0f;
      float lj = -(logf(pos_s) + logf(cp_s) - logf(neg_s)) / cp_s;
      lj = fminf(fmaxf(lj, -100.0f), 100.0f);
      lsum += valid ? lj : 0.0f;
      lcnt += valid ? 1.0f : 0.0f;
    }
    atomicAdd(&acc[0], lsum);
    atomicAdd(&acc[1], lcnt);
  }
}

// ---------------------------------------------------------------- finalize
__global__ void final_kernel(const float* __restrict__ ws, float* __restrict__ out) {
  if (threadIdx.x == 0 && blockIdx.x == 0) {
    float total = ws[0] / fmaxf(ws[1], 1.0f)           // cls
                + ws[2] / fmaxf(ws[3], 1.0f)           // pointer
                + ws[4] / fmaxf(ws[5], 1.0f)           // empty BCE
                + 0.5f * ws[6] / fmaxf(ws[7], 1.0f)    // row contrastive
                + 0.5f * ws[8] / fmaxf(ws[9], 1.0f);   // col contrastive
    out[0] = total;
  }
}

// ---------------------------------------------------------------- launch
extern "C" void kernel_launch(void* const* d_in, const int* in_sizes, int n_in,
                              void* d_out, int out_size, void* d_ws, size_t ws_size,
                              hipStream_t stream) {
  const float* tag    = (const float*)d_in[0];   // [B,L,V]
  const float* pl     = (const float*)d_in[1];   // [B,N,L]
  const float* el     = (const float*)d_in[2];   // [B,L]
  const float* rsm    = (const float*)d_in[3];   // [B,N,N]
  const float* csm    = (const float*)d_in[4];   // [B,N,N]
  const float* rsc    = (const float*)d_in[5];   // [B,N,N]
  const float* csc    = (const float*)d_in[6];   // [B,N,N]
  const int*   tokens = (const int*)d_in[7];     // [B,L]
  const int*   box    = (const int*)d_in[8];     // [B,N,M]
  const int*   mask   = (const int*)d_in[9];     // [B,L] (bool as int)
  float* ws    = (float*)d_ws;                   // [0..9] accumulators
  float* label = ws + 16;                        // [B*L] empty-loss labels
  float* out   = (float*)d_out;

  (void)in_sizes; (void)n_in; (void)out_size; (void)ws_size;

  init_kernel<<<(16 + kB * kL + 255) / 256, 256, 0, stream>>>(ws);
  scatter_kernel<<<(kB * kN * kM + 255) / 256, 256, 0, stream>>>(box, label);
  cls_kernel<<<kB * kL, 256, 0, stream>>>(tag, tokens, ws);
  ptr_kernel<<<kB * kN, 256, 0, stream>>>(pl, box, mask, ws);
  empty_kernel<<<(kB * kL) / 256, 256, 0, stream>>>(el, mask, label, ws);
  span_kernel<<<kB * (kN / 16), 32, 0, stream>>>(rsm, rsc, ws + 6);
  span_kernel<<<kB * (kN / 16), 32, 0, stream>>>(csm, csc, ws + 8);
  final_kernel<<<1, 32, 0, stream>>>(ws, out);
}